// GraphSAGE_SII_16630113370113
// MI455X (gfx1250) — compile-verified
//
#include <hip/hip_runtime.h>
#include <hip/hip_bf16.h>
#include <float.h>

typedef __attribute__((ext_vector_type(2))) float v2f;
typedef __attribute__((ext_vector_type(8))) float v8f;

#define NNODES   50000
#define NEDGES   800000
#define DFEAT    96
#define DINFO    32
#define DEMB     160
#define COUT     128      // conv output dim
#define HCATD    384      // 3 * COUT
#define NGRAPH   500
#define DTGT     10

// ---------------------------------------------------------------- utilities
__global__ void fill_f(float* __restrict__ p, float v, int n) {
    int i = blockIdx.x * blockDim.x + threadIdx.x;
    if (i < n) p[i] = v;
}

// hin0[node, 0:96] = x[node]; hin0[node, 96:128] = struc[batch[node]]
__global__ void build_hin0(const float* __restrict__ x,
                           const float* __restrict__ struc,
                           const int* __restrict__ batch,
                           float* __restrict__ hin0) {
    int node = blockIdx.x;
    int t = threadIdx.x;                       // 128 threads
    float v = (t < DFEAT) ? x[(size_t)node * DFEAT + t]
                          : struc[(size_t)batch[node] * DINFO + (t - DFEAT)];
    hin0[(size_t)node * (DFEAT + DINFO) + t] = v;
}

// hin[node, 128:160] = struc[batch[node]]  (info columns of the 160-wide buffers)
__global__ void fill_info(const float* __restrict__ struc,
                          const int* __restrict__ batch,
                          float* __restrict__ hin) {
    int node = blockIdx.x;
    int t = threadIdx.x;                       // 32 threads
    hin[(size_t)node * DEMB + COUT + t] = struc[(size_t)batch[node] * DINFO + t];
}

__global__ void degree_kernel(const int* __restrict__ dst, float* __restrict__ deg, int E) {
    int e = blockIdx.x * blockDim.x + threadIdx.x;
    if (e < E) atomicAdd(&deg[dst[e]], 1.0f);
}

__global__ void invdeg_kernel(const float* __restrict__ deg, float* __restrict__ idg, int n) {
    int i = blockIdx.x * blockDim.x + threadIdx.x;
    if (i < n) idg[i] = 1.0f / fmaxf(deg[i], 1.0f);
}

// edge-parallel scatter-add: agg[dst] += hin[src], 4 floats per thread
__global__ void scatter_edges(const float* __restrict__ hin,
                              const int* __restrict__ src,
                              const int* __restrict__ dst,
                              float* __restrict__ agg, int din) {
    int chunks = din >> 2;
    long idx = (long)blockIdx.x * blockDim.x + threadIdx.x;
    long total = (long)NEDGES * chunks;
    if (idx >= total) return;
    int e = (int)(idx / chunks);
    int c = (int)(idx % chunks);
    int s = src[e], d = dst[e];
    const float4 v = ((const float4*)(hin + (size_t)s * din))[c];
    float* ap = agg + (size_t)d * din + (size_t)c * 4;
    atomicAdd(ap + 0, v.x);
    atomicAdd(ap + 1, v.y);
    atomicAdd(ap + 2, v.z);
    atomicAdd(ap + 3, v.w);
}

// ---------------------------------------------------------------- WMMA GEMM
// One wave computes a 16x16 tile of:  D = (agg*invdeg) @ Wl + hin @ Wr + bias
// grid = (3125, 8), block = 32 (wave32, EXEC all ones, no tails: 50000=3125*16, 128=8*16)
__global__ __launch_bounds__(32)
void sage_gemm(const float* __restrict__ agg, const float* __restrict__ invdeg,
               const float* __restrict__ hin,
               const float* __restrict__ Wl, const float* __restrict__ Wr,
               const float* __restrict__ bias,
               float* __restrict__ hcat, int hcat_off,
               float* __restrict__ hnext, int din) {
    const int lane = threadIdx.x;
    const int m    = lane & 15;        // A row within tile / B,D column within tile
    const int hi   = lane >> 4;        // second K-pair half
    const int row  = blockIdx.x * 16 + m;
    const int cn   = blockIdx.y * 16;
    const int n    = cn + m;

    const float idg = invdeg[row];
    const float* aggRow = agg + (size_t)row * din;
    const float* hinRow = hin + (size_t)row * din;

    v8f acc = {};
    for (int k = 0; k < din; k += 4) {
        const int ka = k + 2 * hi;     // this lane's K pair: {ka, ka+1}
        v2f a, b;
        // mean @ Wl
        a.x = aggRow[ka]     * idg;
        a.y = aggRow[ka + 1] * idg;
        b.x = Wl[(size_t)ka * COUT + n];
        b.y = Wl[(size_t)(ka + 1) * COUT + n];
        acc = __builtin_amdgcn_wmma_f32_16x16x4_f32(false, a, false, b,
                                                    (short)0, acc, false, false);
        // hin @ Wr
        a.x = hinRow[ka];
        a.y = hinRow[ka + 1];
        b.x = Wr[(size_t)ka * COUT + n];
        b.y = Wr[(size_t)(ka + 1) * COUT + n];
        acc = __builtin_amdgcn_wmma_f32_16x16x4_f32(false, a, false, b,
                                                    (short)0, acc, false, false);
    }

    const float bv = bias[n];
    #pragma unroll
    for (int r = 0; r < 8; ++r) {
        const int rr = blockIdx.x * 16 + r + 8 * hi;   // D: vgpr r -> M=r (+8 for hi lanes)
        const float v = acc[r] + bv;
        hcat[(size_t)rr * HCATD + hcat_off + cn + m] = v;
        if (hnext) hnext[(size_t)rr * DEMB + cn + m] = v;
    }
}

// ---------------------------------------------------------------- pooling + MLP
__device__ __forceinline__ void atomicMaxF(float* addr, float val) {
    int* ai = (int*)addr;
    int old = *ai;
    while (__int_as_float(old) < val) {
        int assumed = old;
        old = atomicCAS(ai, assumed, __float_as_int(val));
        if (old == assumed) break;
    }
}

__global__ void pool_max(const float* __restrict__ hcat,
                         const int* __restrict__ batch,
                         float* __restrict__ g) {
    int node = blockIdx.x;
    int t = threadIdx.x;                       // 384 threads
    atomicMaxF(&g[(size_t)batch[node] * HCATD + t], hcat[(size_t)node * HCATD + t]);
}

__global__ void fc1_relu(const float* __restrict__ g, const float* __restrict__ W,
                         const float* __restrict__ b, float* __restrict__ z) {
    int row = blockIdx.x;                      // 500
    int col = threadIdx.x;                     // 160
    float s = b[col];
    const float* gr = g + (size_t)row * HCATD;
    for (int k = 0; k < HCATD; ++k) s += gr[k] * W[(size_t)k * DEMB + col];
    z[(size_t)row * DEMB + col] = fmaxf(s, 0.0f);
}

__global__ void fc2(const float* __restrict__ z, const float* __restrict__ W,
                    const float* __restrict__ b, float* __restrict__ out) {
    int row = blockIdx.x;                      // 500
    int col = threadIdx.x;                     // 32, guard to 10
    if (col >= DTGT) return;
    float s = b[col];
    const float* zr = z + (size_t)row * DEMB;
    for (int k = 0; k < DEMB; ++k) s += zr[k] * W[(size_t)k * DTGT + col];
    out[(size_t)row * DTGT + col] = s;
}

// ---------------------------------------------------------------- launch
extern "C" void kernel_launch(void* const* d_in, const int* in_sizes, int n_in,
                              void* d_out, int out_size, void* d_ws, size_t ws_size,
                              hipStream_t stream) {
    const float* x     = (const float*)d_in[0];
    const int*   ei    = (const int*)d_in[1];
    const int*   src   = ei;
    const int*   dst   = ei + NEDGES;
    const int*   batch = (const int*)d_in[2];
    const float* struc = (const float*)d_in[3];
    const float* Wl[3] = {(const float*)d_in[4], (const float*)d_in[7], (const float*)d_in[10]};
    const float* Wr[3] = {(const float*)d_in[5], (const float*)d_in[8], (const float*)d_in[11]};
    const float* bb[3] = {(const float*)d_in[6], (const float*)d_in[9], (const float*)d_in[12]};
    const float* Wfc1  = (const float*)d_in[13];
    const float* bfc1  = (const float*)d_in[14];
    const float* Wfc2  = (const float*)d_in[15];
    const float* bfc2  = (const float*)d_in[16];
    float* out = (float*)d_out;

    // workspace carve-up (all fp32)
    float* ws    = (float*)d_ws;
    float* hin0  = ws;                 ws += (size_t)NNODES * (DFEAT + DINFO);  // 128 wide
    float* hinA  = ws;                 ws += (size_t)NNODES * DEMB;             // 160 wide
    float* hinB  = ws;                 ws += (size_t)NNODES * DEMB;
    float* agg   = ws;                 ws += (size_t)NNODES * DEMB;             // max din = 160
    float* deg   = ws;                 ws += (size_t)NNODES;
    float* idg   = ws;                 ws += (size_t)NNODES;
    float* hcat  = ws;                 ws += (size_t)NNODES * HCATD;
    float* g     = ws;                 ws += (size_t)NGRAPH * HCATD;
    float* z     = ws;                 ws += (size_t)NGRAPH * DEMB;

    // inputs for each conv layer
    build_hin0<<<NNODES, 128, 0, stream>>>(x, struc, batch, hin0);
    fill_info<<<NNODES, 32, 0, stream>>>(struc, batch, hinA);
    fill_info<<<NNODES, 32, 0, stream>>>(struc, batch, hinB);

    // degrees (same every layer)
    fill_f<<<(NNODES + 255) / 256, 256, 0, stream>>>(deg, 0.0f, NNODES);
    degree_kernel<<<(NEDGES + 255) / 256, 256, 0, stream>>>(dst, deg, NEDGES);
    invdeg_kernel<<<(NNODES + 255) / 256, 256, 0, stream>>>(deg, idg, NNODES);

    const float* hins[3]  = {hin0, hinA, hinB};
    float*       hnext[3] = {hinA, hinB, nullptr};
    const int    dins[3]  = {DFEAT + DINFO, DEMB, DEMB};

    for (int l = 0; l < 3; ++l) {
        const int din = dins[l];
        const int aggN = NNODES * din;
        fill_f<<<(aggN + 255) / 256, 256, 0, stream>>>(agg, 0.0f, aggN);
        const long total = (long)NEDGES * (din >> 2);
        scatter_edges<<<(int)((total + 255) / 256), 256, 0, stream>>>(hins[l], src, dst, agg, din);
        dim3 grid(NNODES / 16, COUT / 16);     // (3125, 8), exact tiling
        sage_gemm<<<grid, 32, 0, stream>>>(agg, idg, hins[l], Wl[l], Wr[l], bb[l],
                                           hcat, l * COUT, hnext[l], din);
    }

    // global max pool + MLP head
    fill_f<<<(NGRAPH * HCATD + 255) / 256, 256, 0, stream>>>(g, -FLT_MAX, NGRAPH * HCATD);
    pool_max<<<NNODES, HCATD, 0, stream>>>(hcat, batch, g);
    fc1_relu<<<NGRAPH, DEMB, 0, stream>>>(g, Wfc1, bfc1, z);
    fc2<<<NGRAPH, 32, 0, stream>>>(z, Wfc2, bfc2, out);
}